// MSSIM_24842090840526
// MI455X (gfx1250) — compile-verified
//
#include <hip/hip_runtime.h>
#include <hip/hip_bf16.h>
#include <math.h>

// ---------------------------------------------------------------------------
// MS-SSIM (AntixK/PyTorch-VAE faithful, incl. positive-exponent Gaussian bug)
// fp32 separable 11x11 depthwise conv through LDS, WMMA f32 16x16x4 ones-trick
// block/global reductions (deterministic, no float atomics).
// TH=32 tiles: vertical-halo overwork for the horizontal pass drops from
// 1.625x to 1.31x (~13% less VALU work in the dominant kernel).
// ---------------------------------------------------------------------------

typedef float v2f __attribute__((ext_vector_type(2)));
typedef float v8f __attribute__((ext_vector_type(8)));

#define KW   11
#define PADV 5
#define TW   64
#define TH   32
#define RW   (TW + 2 * PADV)   // 74
#define RH   (TH + 2 * PADV)   // 42

// Sum 256 fp32 values (in LDS) using V_WMMA_F32_16X16X4_F32 twice:
//   S1 = M x Ones  (row sums, replicated along columns)
//   S2 = Ones x S1 (total sum in every element)
// Total-sum is invariant to the exact A/B lane striping: every input element
// is consumed exactly once and the ones operands are layout-independent.
// Must be executed by one full wave (lanes 0..31, EXEC all ones).
__device__ __forceinline__ float wmma_sum256(const float* __restrict__ m, int lane) {
  v2f ones; ones.x = 1.0f; ones.y = 1.0f;
  v8f acc = {};
#pragma unroll
  for (int kb = 0; kb < 4; ++kb) {
    v2f a;
    a.x = m[(2 * kb + 0) * 32 + lane];
    a.y = m[(2 * kb + 1) * 32 + lane];
    acc = __builtin_amdgcn_wmma_f32_16x16x4_f32(false, a, false, ones,
                                                (short)0, acc, false, false);
  }
  v8f acc2 = {};
#pragma unroll
  for (int kb = 0; kb < 4; ++kb) {
    v2f b;
    b.x = acc[2 * kb + 0];
    b.y = acc[2 * kb + 1];
    acc2 = __builtin_amdgcn_wmma_f32_16x16x4_f32(false, ones, false, b,
                                                 (short)0, acc2, false, false);
  }
  return acc2[0];
}

// Per-scale SSIM statistics: separable Gaussian conv of {i1,i2,i1^2,i2^2,i1*i2}
// through LDS, per-pixel ssim_map / cs values, per-block partial sums via WMMA.
__global__ __launch_bounds__(256) void ssim_stats_kernel(
    const float* __restrict__ im1, const float* __restrict__ im2,
    int H, int W,
    float* __restrict__ part_ssim, float* __restrict__ part_cs) {
  __shared__ float g[KW];
  __shared__ float r1[RH][RW];
  __shared__ float r2[RH][RW];
  __shared__ float h1 [RH][TW];
  __shared__ float h2 [RH][TW];
  __shared__ float h11[RH][TW];
  __shared__ float h22[RH][TW];
  __shared__ float h12[RH][TW];
  __shared__ float red[512];

  const int tid = threadIdx.x;

  // Gaussian taps in fp64 (replicates numpy float64 math, POSITIVE exponent).
  if (tid == 0) {
    double kk[KW];
    double s = 0.0;
    for (int i = 0; i < KW; ++i) {
      double d = (double)(i - PADV);
      kk[i] = exp(d * d / (2.0 * 1.5 * 1.5));
      s += kk[i];
    }
    for (int i = 0; i < KW; ++i) g[i] = (float)(kk[i] / s);
  }

  const int img = blockIdx.z;          // 0 .. B*C-1 (96)
  const int r0  = blockIdx.y * TH;
  const int c0  = blockIdx.x * TW;
  const float* __restrict__ p1 = im1 + (size_t)img * H * W;
  const float* __restrict__ p2 = im2 + (size_t)img * H * W;

  // Phase 1: stage raw tile + halo (zero-padded) into LDS.
  for (int idx = tid; idx < RH * RW; idx += 256) {
    const int rr = idx / RW, cc = idx % RW;
    const int gr = r0 + rr - PADV, gc = c0 + cc - PADV;
    float a = 0.0f, b = 0.0f;
    if (gr >= 0 && gr < H && gc >= 0 && gc < W) {
      a = p1[(size_t)gr * W + gc];
      b = p2[(size_t)gr * W + gc];
    }
    r1[rr][cc] = a;
    r2[rr][cc] = b;
  }
  __syncthreads();

  // Phase 2: horizontal 11-tap conv of the 5 quantities.
  for (int idx = tid; idx < RH * TW; idx += 256) {
    const int rr = idx / TW, cc = idx % TW;
    float a1 = 0.f, a2 = 0.f, a11 = 0.f, a22 = 0.f, a12 = 0.f;
#pragma unroll
    for (int t = 0; t < KW; ++t) {
      const float w = g[t];
      const float x = r1[rr][cc + t];
      const float y = r2[rr][cc + t];
      a1  += w * x;
      a2  += w * y;
      a11 += w * x * x;
      a22 += w * y * y;
      a12 += w * x * y;
    }
    h1 [rr][cc] = a1;
    h2 [rr][cc] = a2;
    h11[rr][cc] = a11;
    h22[rr][cc] = a22;
    h12[rr][cc] = a12;
  }
  __syncthreads();

  // Phase 3: vertical 11-tap conv + per-pixel SSIM terms.
  const float C1 = 1.0e-4f;   // 0.01^2
  const float C2 = 9.0e-4f;   // 0.03^2
  float ps = 0.0f, pc = 0.0f;
  for (int idx = tid; idx < TH * TW; idx += 256) {
    const int ry = idx / TW, cx = idx % TW;
    const int gr = r0 + ry, gc = c0 + cx;
    if (gr < H && gc < W) {
      float mu1 = 0.f, mu2 = 0.f, m11 = 0.f, m22 = 0.f, m12 = 0.f;
#pragma unroll
      for (int t = 0; t < KW; ++t) {
        const float w = g[t];
        mu1 += w * h1 [ry + t][cx];
        mu2 += w * h2 [ry + t][cx];
        m11 += w * h11[ry + t][cx];
        m22 += w * h22[ry + t][cx];
        m12 += w * h12[ry + t][cx];
      }
      const float mu1s = mu1 * mu1;
      const float mu2s = mu2 * mu2;
      const float mu12 = mu1 * mu2;
      const float s1  = m11 - mu1s;
      const float s2v = m22 - mu2s;
      const float s12 = m12 - mu12;
      const float v1 = 2.0f * s12 + C2;
      const float v2 = s1 + s2v + C2;
      pc += v1 / v2;
      ps += ((2.0f * mu12 + C1) * v1) / ((mu1s + mu2s + C1) * v2);
    }
  }

  // Phase 4: block reduction 256 -> 1 via WMMA ones-trick (wave 0).
  red[tid]       = ps;
  red[256 + tid] = pc;
  __syncthreads();
  if (tid < 32) {
    const float ts = wmma_sum256(&red[0],   tid);
    const float tc = wmma_sum256(&red[256], tid);
    if (tid == 0) {
      const int bid = (blockIdx.z * gridDim.y + blockIdx.y) * gridDim.x + blockIdx.x;
      part_ssim[bid] = ts;
      part_cs[bid]   = tc;
    }
  }
}

// Deterministic fixed-order reduction of per-block partials -> per-scale means.
__global__ __launch_bounds__(256) void reduce_stats_kernel(
    const float* __restrict__ part_ssim, const float* __restrict__ part_cs,
    int n, float inv_count, float* __restrict__ out2) {
  __shared__ float red[512];
  const int tid = threadIdx.x;
  float s = 0.0f, c = 0.0f;
  for (int i = tid; i < n; i += 256) {
    s += part_ssim[i];
    c += part_cs[i];
  }
  red[tid]       = s;
  red[256 + tid] = c;
  __syncthreads();
  if (tid < 32) {
    const float ts = wmma_sum256(&red[0],   tid);
    const float tc = wmma_sum256(&red[256], tid);
    if (tid == 0) {
      out2[0] = ts * inv_count;   // mssim at this scale
      out2[1] = tc * inv_count;   // mcs   at this scale
    }
  }
}

// 2x2 average pool (pyramid downsample).
__global__ __launch_bounds__(256) void avgpool2_kernel(
    const float* __restrict__ in, float* __restrict__ out, int Ho, int Wo) {
  const int i = blockIdx.x * blockDim.x + threadIdx.x;
  const int total = 96 * Ho * Wo;
  if (i >= total) return;
  const int wo  = i % Wo;
  const int t   = i / Wo;
  const int ho  = t % Ho;
  const int img = t / Ho;
  const int Wi = Wo * 2, Hi = Ho * 2;
  const float* __restrict__ p = in + (size_t)img * Hi * Wi;
  const float v = 0.25f * (p[(size_t)(2 * ho) * Wi + 2 * wo] +
                           p[(size_t)(2 * ho) * Wi + 2 * wo + 1] +
                           p[(size_t)(2 * ho + 1) * Wi + 2 * wo] +
                           p[(size_t)(2 * ho + 1) * Wi + 2 * wo + 1]);
  out[i] = v;
}

// output = 1 - prod(mcs[0..3]^w[i]) * (mssim[4]^w[4])^4   (faithful to repo)
__global__ void finalize_kernel(const float* __restrict__ accum,
                                float* __restrict__ out) {
  if (threadIdx.x == 0 && blockIdx.x == 0) {
    const float w[5] = {0.0448f, 0.2856f, 0.3001f, 0.2363f, 0.1333f};
    float prod = 1.0f;
#pragma unroll
    for (int i = 0; i < 4; ++i) prod *= powf(accum[2 * i + 1], w[i]);  // mcs_i
    prod *= powf(accum[8], 4.0f * w[4]);                               // mssim_4
    out[0] = 1.0f - prod;
  }
}

extern "C" void kernel_launch(void* const* d_in, const int* in_sizes, int n_in,
                              void* d_out, int out_size, void* d_ws, size_t ws_size,
                              hipStream_t stream) {
  (void)in_sizes; (void)n_in; (void)out_size; (void)ws_size;
  const float* img1 = (const float*)d_in[0];
  const float* img2 = (const float*)d_in[1];
  float* out = (float*)d_out;
  float* ws  = (float*)d_ws;

  // Workspace layout (floats):
  //   [0..15]                       accum (mssim/mcs pairs, 10 used)
  //   [16 .. 16+PARTMAX)            per-block partial ssim sums
  //   [.. +PARTMAX)                 per-block partial cs sums
  //   then: img1 pyramid s1..s4, img2 pyramid s1..s4
  const int PARTMAX = 16384;  // scale-0 block count: 8 x 16 x 96 = 12288
  float* accum  = ws;
  float* part_s = ws + 16;
  float* part_c = part_s + PARTMAX;
  float* pyr    = part_c + PARTMAX;

  // Pyramid offsets for scales 1..4 (96 images per level).
  size_t lvl_off[5];
  size_t cum = 0;
  lvl_off[0] = 0;  // unused (scale 0 reads d_in)
  for (int s = 1; s <= 4; ++s) {
    lvl_off[s] = cum;
    const int hs = 512 >> s;
    cum += (size_t)96 * hs * hs;
  }
  const size_t pyr_per_img = cum;
  float* p1lvl[5]; float* p2lvl[5];
  for (int s = 1; s <= 4; ++s) {
    p1lvl[s] = pyr + lvl_off[s];
    p2lvl[s] = pyr + pyr_per_img + lvl_off[s];
  }

  for (int s = 0; s < 5; ++s) {
    const int H = 512 >> s;
    const int W = H;
    const float* src1 = (s == 0) ? img1 : p1lvl[s];
    const float* src2 = (s == 0) ? img2 : p2lvl[s];

    const int gx = (W + TW - 1) / TW;
    const int gy = (H + TH - 1) / TH;
    dim3 grid(gx, gy, 96);
    ssim_stats_kernel<<<grid, 256, 0, stream>>>(src1, src2, H, W, part_s, part_c);

    const int nblocks = gx * gy * 96;
    const float inv_count = 1.0f / (96.0f * (float)H * (float)W);
    reduce_stats_kernel<<<1, 256, 0, stream>>>(part_s, part_c, nblocks,
                                               inv_count, accum + 2 * s);

    if (s < 4) {
      const int Ho = H / 2, Wo = W / 2;
      const int total = 96 * Ho * Wo;
      const int nb = (total + 255) / 256;
      avgpool2_kernel<<<nb, 256, 0, stream>>>(src1, p1lvl[s + 1], Ho, Wo);
      avgpool2_kernel<<<nb, 256, 0, stream>>>(src2, p2lvl[s + 1], Ho, Wo);
    }
  }

  finalize_kernel<<<1, 64, 0, stream>>>(accum, out);
}